// MambaDecoderLayerWithCross_28965259444908
// MI455X (gfx1250) — compile-verified
//
#include <hip/hip_runtime.h>
#include <math.h>

// ---------------------------------------------------------------------------
// MambaDecoderLayerWithCross for MI455X (gfx1250, wave32, WMMA)
// All dense projections use V_WMMA_F32_16X16X4_F32 (fp32-exact matrix path).
// Register-blocked 16 x (16*NT) strip per wave + explicit double buffering:
// fragments for k+4 are loaded while the k WMMAs execute, so waits before a
// WMMA group only cover loads issued one full iteration earlier.
// Selective scan: one wave per (b, d_inner) channel, 4 states/lane, shfl reduce.
// ---------------------------------------------------------------------------

typedef float v2f __attribute__((ext_vector_type(2)));
typedef float v8f __attribute__((ext_vector_type(8)));

#define D_MODEL 256
#define D_STATE 128
#define D_CONV  4
#define D_INNER 1024
#define DT_RANK 16
#define N_HEADS 8
#define D_FF    1024
#define BATCH   2
#define SEQ     512
#define TOKENS  (BATCH * SEQ)       // 1024
#define DK      (D_MODEL / N_HEADS) // 32
#define DBL_N   (DT_RANK + 2 * D_STATE) // 272
#define LN_EPS  1e-6f
#define NEG_INF (-3.402823466e38f)

__device__ __forceinline__ float sigmoidf_(float x) { return 1.0f / (1.0f + __expf(-x)); }
__device__ __forceinline__ float siluf_(float x)    { return x * sigmoidf_(x); }
__device__ __forceinline__ float softplusf_(float x){ return (x > 20.0f) ? x : log1pf(expf(x)); }

// ------------------------------ WMMA GEMM ----------------------------------
// C[m,n] = alpha * sum_k A[m,k] * B(k,n)  (+ epilogue)
//   BL_NT: B stored (N,K) row-major  -> B(k,n) = Bm[n*ldb + k]   (weights, Q@K^T)
//   BL_NN: B stored (K,N) row-major  -> B(k,n) = Bm[k*ldb + n]   (P@V)
// Fragment layouts per CDNA5 ISA 7.12.2:
//   A 16x4 f32 : lane l, vgpr v -> M = l%16, K = v + 2*(l/16)
//   B  4x16 f32: lane l, vgpr v -> N = l%16, K = v + 2*(l/16)
//   C 16x16 f32: lane l, vgpr v -> N = l%16, M = v + 8*(l/16)

enum { BL_NT = 0, BL_NN = 1 };
enum { EP_NONE = 0, EP_BIAS, EP_BIAS_RELU, EP_BIAS_RESID, EP_RESID,
       EP_SOFTPLUS_BIAS, EP_SCALE_MASK };

struct GemmArgs {
  const float* A;  int lda;  long sA1, sA2;
  const float* Bm; int ldb;  long sB1, sB2;
  float*       C;  int ldc;  long sC1, sC2;
  int M, N, K, H;
  const float* bias;
  const float* resid; int ldr;
  float alpha;
  const unsigned char* mask; long maskStride;
};

template <int BLAYOUT, int EPI, int NTILES>
__global__ __launch_bounds__(32) void wmma_gemm_kernel(GemmArgs g) {
  const int lane = threadIdx.x & 31;
  const int half = lane >> 4;        // 0..1
  const int lr   = lane & 15;        // 0..15
  const int n0   = blockIdx.x * 16 * NTILES;
  const int m0   = blockIdx.y * 16;
  const int z    = blockIdx.z;
  const int zb   = z / g.H;
  const int zh   = z % g.H;

  const float* A  = g.A  + zb * g.sA1 + zh * g.sA2;
  const float* Bp = g.Bm + zb * g.sB1 + zh * g.sB2;
  float*       C  = g.C  + zb * g.sC1 + zh * g.sC2;

  v8f acc[NTILES];
#pragma unroll
  for (int t = 0; t < NTILES; ++t) acc[t] = (v8f){};

  const float* arow = A + (long)(m0 + lr) * g.lda + 2 * half;

  if (BLAYOUT == BL_NT) {
    const float* brow[NTILES];
#pragma unroll
    for (int t = 0; t < NTILES; ++t)
      brow[t] = Bp + (long)(n0 + 16 * t + lr) * g.ldb + 2 * half;

    // preload k = 0
    v2f a = *(const v2f*)(arow);
    v2f b[NTILES];
#pragma unroll
    for (int t = 0; t < NTILES; ++t) b[t] = *(const v2f*)(brow[t]);

    int k = 0;
    for (; k + 4 < g.K; k += 4) {
      // prefetch k+4 fragments (independent of current WMMA group)
      v2f an = *(const v2f*)(arow + k + 4);
      v2f bn[NTILES];
#pragma unroll
      for (int t = 0; t < NTILES; ++t) bn[t] = *(const v2f*)(brow[t] + k + 4);
      // WMMA group on current fragments
#pragma unroll
      for (int t = 0; t < NTILES; ++t)
        acc[t] = __builtin_amdgcn_wmma_f32_16x16x4_f32(false, a, false, b[t],
                                                       (short)0, acc[t], false, false);
      a = an;
#pragma unroll
      for (int t = 0; t < NTILES; ++t) b[t] = bn[t];
    }
    // final k-step (no prefetch)
#pragma unroll
    for (int t = 0; t < NTILES; ++t)
      acc[t] = __builtin_amdgcn_wmma_f32_16x16x4_f32(false, a, false, b[t],
                                                     (short)0, acc[t], false, false);
  } else {
    // preload k = 0
    v2f a = *(const v2f*)(arow);
    v2f b[NTILES];
#pragma unroll
    for (int t = 0; t < NTILES; ++t) {
      b[t].x = Bp[(long)(2 * half + 0) * g.ldb + n0 + 16 * t + lr];
      b[t].y = Bp[(long)(2 * half + 1) * g.ldb + n0 + 16 * t + lr];
    }
    int k = 0;
    for (; k + 4 < g.K; k += 4) {
      const int ka = k + 4 + 2 * half;
      v2f an = *(const v2f*)(arow + k + 4);
      v2f bn[NTILES];
#pragma unroll
      for (int t = 0; t < NTILES; ++t) {
        bn[t].x = Bp[(long)(ka + 0) * g.ldb + n0 + 16 * t + lr];
        bn[t].y = Bp[(long)(ka + 1) * g.ldb + n0 + 16 * t + lr];
      }
#pragma unroll
      for (int t = 0; t < NTILES; ++t)
        acc[t] = __builtin_amdgcn_wmma_f32_16x16x4_f32(false, a, false, b[t],
                                                       (short)0, acc[t], false, false);
      a = an;
#pragma unroll
      for (int t = 0; t < NTILES; ++t) b[t] = bn[t];
    }
#pragma unroll
    for (int t = 0; t < NTILES; ++t)
      acc[t] = __builtin_amdgcn_wmma_f32_16x16x4_f32(false, a, false, b[t],
                                                     (short)0, acc[t], false, false);
  }

#pragma unroll
  for (int t = 0; t < NTILES; ++t) {
    const int n = n0 + 16 * t + lr;
    float bias_n = 0.0f;
    if constexpr (EPI == EP_BIAS || EPI == EP_BIAS_RELU ||
                  EPI == EP_BIAS_RESID || EPI == EP_SOFTPLUS_BIAS)
      bias_n = g.bias[n];
    unsigned char mk = 1;
    if constexpr (EPI == EP_SCALE_MASK)
      mk = g.mask[zb * g.maskStride + n];

#pragma unroll
    for (int v = 0; v < 8; ++v) {
      const int m = m0 + v + 8 * half;
      float val = acc[t][v];
      if constexpr (EPI == EP_SCALE_MASK) {
        val *= g.alpha;
        if (!mk) val = NEG_INF;
      }
      if constexpr (EPI == EP_BIAS || EPI == EP_BIAS_RELU ||
                    EPI == EP_BIAS_RESID || EPI == EP_SOFTPLUS_BIAS)
        val += bias_n;
      if constexpr (EPI == EP_BIAS_RELU)     val = fmaxf(val, 0.0f);
      if constexpr (EPI == EP_SOFTPLUS_BIAS) val = softplusf_(val);
      if constexpr (EPI == EP_BIAS_RESID || EPI == EP_RESID)
        val += g.resid[(long)m * g.ldr + n];
      C[(long)m * g.ldc + n] = val;
    }
  }
}

// ------------------------------ LayerNorm ----------------------------------
// One block (256 threads) per token row; reference uses (d-1) variance divisor.
__global__ __launch_bounds__(D_MODEL) void layernorm_kernel(
    const float* __restrict__ x, const float* __restrict__ a,
    const float* __restrict__ b, float* __restrict__ y) {
  __shared__ float red[D_MODEL];
  const long row = blockIdx.x;
  const int  t   = threadIdx.x;
  const float v  = x[row * D_MODEL + t];
  red[t] = v; __syncthreads();
  for (int s = D_MODEL / 2; s > 0; s >>= 1) {
    if (t < s) red[t] += red[t + s];
    __syncthreads();
  }
  const float mean = red[0] * (1.0f / D_MODEL);
  __syncthreads();
  const float d = v - mean;
  red[t] = d * d; __syncthreads();
  for (int s = D_MODEL / 2; s > 0; s >>= 1) {
    if (t < s) red[t] += red[t + s];
    __syncthreads();
  }
  const float var = red[0] * (1.0f / (D_MODEL - 1));
  y[row * D_MODEL + t] = a[t] * d / (sqrtf(var) + LN_EPS) + b[t];
}

// ----------------------- Causal depthwise conv4 + SiLU ---------------------
__global__ void conv_silu_kernel(const float* __restrict__ xz,
                                 const float* __restrict__ cw,
                                 const float* __restrict__ cb,
                                 float* __restrict__ u) {
  const int idx = blockIdx.x * blockDim.x + threadIdx.x;
  if (idx >= TOKENS * D_INNER) return;
  const int d = idx % D_INNER;
  const int t = (idx / D_INNER) % SEQ;
  const int b = idx / (D_INNER * SEQ);
  float acc = cb[d];
#pragma unroll
  for (int k = 0; k < D_CONV; ++k) {
    const int tt = t + k - (D_CONV - 1);
    if (tt >= 0)
      acc += xz[((long)(b * SEQ + tt)) * (2 * D_INNER) + d] * cw[d * D_CONV + k];
  }
  u[idx] = siluf_(acc);
}

// ---------------------- Selective scan (one wave per (b,d)) ----------------
// h[n] <- exp(dt*A[d,n])*h[n] + dt*B[t,n]*u ; y_t = sum_n h[n]*C[t,n]
// 4 states per lane (128 states / wave32); shfl_xor reduction for y.
// Fused: y = (y + u*Dp[d]) * silu(z) written to yg.
__global__ __launch_bounds__(256) void scan_kernel(
    const float* __restrict__ dts, const float* __restrict__ u,
    const float* __restrict__ dbl, const float* __restrict__ A_log,
    const float* __restrict__ Dp,  const float* __restrict__ xz,
    float* __restrict__ yg) {
  const int wid  = (blockIdx.x * blockDim.x + threadIdx.x) >> 5;
  const int lane = threadIdx.x & 31;
  if (wid >= BATCH * D_INNER) return;
  const int b  = wid / D_INNER;
  const int d  = wid % D_INNER;
  const int n0 = lane * 4;

  float A0[4], h[4];
#pragma unroll
  for (int i = 0; i < 4; ++i) {
    A0[i] = -__expf(A_log[(long)d * D_STATE + n0 + i]);
    h[i]  = 0.0f;
  }
  const float Dpd = Dp[d];

  for (int t = 0; t < SEQ; ++t) {
    const long tok = (long)b * SEQ + t;
    const float dt = dts[tok * D_INNER + d];
    const float uv = u[tok * D_INNER + d];
    const float4 Bv = *(const float4*)(dbl + tok * DBL_N + DT_RANK + n0);
    const float4 Cv = *(const float4*)(dbl + tok * DBL_N + DT_RANK + D_STATE + n0);
    const float dtu = dt * uv;
    float part;
    h[0] = __expf(dt * A0[0]) * h[0] + dtu * Bv.x; part  = h[0] * Cv.x;
    h[1] = __expf(dt * A0[1]) * h[1] + dtu * Bv.y; part += h[1] * Cv.y;
    h[2] = __expf(dt * A0[2]) * h[2] + dtu * Bv.z; part += h[2] * Cv.z;
    h[3] = __expf(dt * A0[3]) * h[3] + dtu * Bv.w; part += h[3] * Cv.w;
#pragma unroll
    for (int off = 16; off > 0; off >>= 1)
      part += __shfl_xor(part, off, 32);
    if (lane == 0) {
      const float y = part + uv * Dpd;
      const float z = xz[tok * (2 * D_INNER) + D_INNER + d];
      yg[tok * D_INNER + d] = y * siluf_(z);
    }
  }
}

// ------------------------------ Softmax (rows) -----------------------------
__global__ __launch_bounds__(128) void softmax_kernel(float* __restrict__ s, int cols) {
  __shared__ float red[128];
  const long row = blockIdx.x;
  float* p = s + row * cols;
  const int t = threadIdx.x;
  float mx = NEG_INF;
  for (int c = t; c < cols; c += 128) mx = fmaxf(mx, p[c]);
  red[t] = mx; __syncthreads();
  for (int st = 64; st > 0; st >>= 1) {
    if (t < st) red[t] = fmaxf(red[t], red[t + st]);
    __syncthreads();
  }
  mx = red[0]; __syncthreads();
  float sum = 0.0f;
  for (int c = t; c < cols; c += 128) {
    const float e = __expf(p[c] - mx);
    p[c] = e; sum += e;
  }
  red[t] = sum; __syncthreads();
  for (int st = 64; st > 0; st >>= 1) {
    if (t < st) red[t] += red[t + st];
    __syncthreads();
  }
  const float inv = 1.0f / red[0];
  for (int c = t; c < cols; c += 128) p[c] *= inv;
}

// ------------------------------ host side ----------------------------------
static inline GemmArgs mkargs(const float* A, int lda, const float* Bm, int ldb,
                              float* C, int ldc, int M, int N, int K) {
  GemmArgs g{};
  g.A = A; g.lda = lda; g.sA1 = 0; g.sA2 = 0;
  g.Bm = Bm; g.ldb = ldb; g.sB1 = 0; g.sB2 = 0;
  g.C = C; g.ldc = ldc; g.sC1 = 0; g.sC2 = 0;
  g.M = M; g.N = N; g.K = K; g.H = 1;
  g.bias = nullptr; g.resid = nullptr; g.ldr = ldc;
  g.alpha = 1.0f; g.mask = nullptr; g.maskStride = 0;
  return g;
}

extern "C" void kernel_launch(void* const* d_in, const int* in_sizes, int n_in,
                              void* d_out, int out_size, void* d_ws, size_t ws_size,
                              hipStream_t stream) {
  (void)in_sizes; (void)n_in; (void)out_size; (void)ws_size;
  const float* x         = (const float*)d_in[0];
  const float* memory    = (const float*)d_in[1];
  const unsigned char* src_mask = (const unsigned char*)d_in[2]; // (B,1,Lm) bool
  const float* in_proj_w = (const float*)d_in[4];
  const float* conv_w    = (const float*)d_in[5];
  const float* conv_b    = (const float*)d_in[6];
  const float* x_proj_w  = (const float*)d_in[7];
  const float* dt_proj_w = (const float*)d_in[8];
  const float* dt_proj_b = (const float*)d_in[9];
  const float* A_log     = (const float*)d_in[10];
  const float* Dp        = (const float*)d_in[11];
  const float* out_proj_w= (const float*)d_in[12];
  const float* n1a = (const float*)d_in[13], *n1b = (const float*)d_in[14];
  const float* n2a = (const float*)d_in[15], *n2b = (const float*)d_in[16];
  const float* n3a = (const float*)d_in[17], *n3b = (const float*)d_in[18];
  const float* wq_w = (const float*)d_in[19], *wq_b = (const float*)d_in[20];
  const float* wk_w = (const float*)d_in[21], *wk_b = (const float*)d_in[22];
  const float* wv_w = (const float*)d_in[23], *wv_b = (const float*)d_in[24];
  const float* wo_w = (const float*)d_in[25], *wo_b = (const float*)d_in[26];
  const float* w1_w = (const float*)d_in[27], *w1_b = (const float*)d_in[28];
  const float* w2_w = (const float*)d_in[29], *w2_b = (const float*)d_in[30];
  float* out = (float*)d_out;

  // workspace carve-up (floats)
  float* ws = (float*)d_ws;
  size_t off = 0;
  auto alloc = [&](size_t n) { float* p = ws + off; off += n; return p; };
  float* xn     = alloc((size_t)TOKENS * D_MODEL);
  float* xz     = alloc((size_t)TOKENS * 2 * D_INNER);
  float* u      = alloc((size_t)TOKENS * D_INNER);
  float* dbl    = alloc((size_t)TOKENS * DBL_N);
  float* dts    = alloc((size_t)TOKENS * D_INNER);
  float* yg     = alloc((size_t)TOKENS * D_INNER);
  float* h1     = alloc((size_t)TOKENS * D_MODEL);
  float* hn2    = alloc((size_t)TOKENS * D_MODEL);
  float* q      = alloc((size_t)TOKENS * D_MODEL);
  float* kk     = alloc((size_t)TOKENS * D_MODEL);
  float* vv     = alloc((size_t)TOKENS * D_MODEL);
  float* scores = alloc((size_t)BATCH * N_HEADS * SEQ * SEQ);
  float* o      = alloc((size_t)TOKENS * D_MODEL);
  float* h2     = alloc((size_t)TOKENS * D_MODEL);
  float* hn3    = alloc((size_t)TOKENS * D_MODEL);
  float* ff1    = alloc((size_t)TOKENS * D_FF);

  const dim3 blk32(32);

  // 1) norm1
  layernorm_kernel<<<TOKENS, D_MODEL, 0, stream>>>(x, n1a, n1b, xn);

  // 2) in_proj: xz = xn(1024x256) * in_proj_w(2048x256)^T   [NT=4 strip]
  {
    GemmArgs g = mkargs(xn, D_MODEL, in_proj_w, D_MODEL, xz, 2 * D_INNER,
                        TOKENS, 2 * D_INNER, D_MODEL);
    wmma_gemm_kernel<BL_NT, EP_NONE, 4><<<dim3(2 * D_INNER / 64, TOKENS / 16, 1), blk32, 0, stream>>>(g);
  }

  // 3) depthwise conv + silu -> u
  conv_silu_kernel<<<(TOKENS * D_INNER + 255) / 256, 256, 0, stream>>>(xz, conv_w, conv_b, u);

  // 4) x_proj: dbl = u(1024x1024) * x_proj_w(272x1024)^T  (N=272=17*16 -> NT=1)
  {
    GemmArgs g = mkargs(u, D_INNER, x_proj_w, D_INNER, dbl, DBL_N,
                        TOKENS, DBL_N, D_INNER);
    wmma_gemm_kernel<BL_NT, EP_NONE, 1><<<dim3(DBL_N / 16, TOKENS / 16, 1), blk32, 0, stream>>>(g);
  }

  // 5) dt_proj + softplus: dts = softplus(dbl[:, :16] * dt_proj_w(1024x16)^T + b)
  {
    GemmArgs g = mkargs(dbl, DBL_N, dt_proj_w, DT_RANK, dts, D_INNER,
                        TOKENS, D_INNER, DT_RANK);
    g.bias = dt_proj_b;
    wmma_gemm_kernel<BL_NT, EP_SOFTPLUS_BIAS, 4><<<dim3(D_INNER / 64, TOKENS / 16, 1), blk32, 0, stream>>>(g);
  }

  // 6) selective scan (+u*Dp, *silu(z)) -> yg
  scan_kernel<<<(BATCH * D_INNER * 32) / 256, 256, 0, stream>>>(dts, u, dbl, A_log, Dp, xz, yg);

  // 7) out_proj + residual x: h1 = yg * out_proj_w(256x1024)^T + x
  {
    GemmArgs g = mkargs(yg, D_INNER, out_proj_w, D_INNER, h1, D_MODEL,
                        TOKENS, D_MODEL, D_INNER);
    g.resid = x; g.ldr = D_MODEL;
    wmma_gemm_kernel<BL_NT, EP_RESID, 4><<<dim3(D_MODEL / 64, TOKENS / 16, 1), blk32, 0, stream>>>(g);
  }

  // 8) norm2
  layernorm_kernel<<<TOKENS, D_MODEL, 0, stream>>>(h1, n2a, n2b, hn2);

  // 9) q/k/v projections (kept in (B, L, H*dk) layout)
  {
    GemmArgs g = mkargs(hn2, D_MODEL, wq_w, D_MODEL, q, D_MODEL, TOKENS, D_MODEL, D_MODEL);
    g.bias = wq_b;
    wmma_gemm_kernel<BL_NT, EP_BIAS, 4><<<dim3(D_MODEL / 64, TOKENS / 16, 1), blk32, 0, stream>>>(g);
  }
  {
    GemmArgs g = mkargs(memory, D_MODEL, wk_w, D_MODEL, kk, D_MODEL, TOKENS, D_MODEL, D_MODEL);
    g.bias = wk_b;
    wmma_gemm_kernel<BL_NT, EP_BIAS, 4><<<dim3(D_MODEL / 64, TOKENS / 16, 1), blk32, 0, stream>>>(g);
  }
  {
    GemmArgs g = mkargs(memory, D_MODEL, wv_w, D_MODEL, vv, D_MODEL, TOKENS, D_MODEL, D_MODEL);
    g.bias = wv_b;
    wmma_gemm_kernel<BL_NT, EP_BIAS, 4><<<dim3(D_MODEL / 64, TOKENS / 16, 1), blk32, 0, stream>>>(g);
  }

  // 10) scores = q @ k^T / sqrt(dk), masked; batched over (b,h)
  {
    GemmArgs g{};
    g.A = q;  g.lda = D_MODEL; g.sA1 = (long)SEQ * D_MODEL; g.sA2 = DK;
    g.Bm = kk; g.ldb = D_MODEL; g.sB1 = (long)SEQ * D_MODEL; g.sB2 = DK;
    g.C = scores; g.ldc = SEQ; g.sC1 = (long)N_HEADS * SEQ * SEQ; g.sC2 = (long)SEQ * SEQ;
    g.M = SEQ; g.N = SEQ; g.K = DK; g.H = N_HEADS;
    g.alpha = 1.0f / sqrtf((float)DK);
    g.mask = src_mask; g.maskStride = SEQ;
    g.bias = nullptr; g.resid = nullptr; g.ldr = SEQ;
    wmma_gemm_kernel<BL_NT, EP_SCALE_MASK, 4><<<dim3(SEQ / 64, SEQ / 16, BATCH * N_HEADS), blk32, 0, stream>>>(g);
  }

  // 11) softmax rows (in place)
  softmax_kernel<<<BATCH * N_HEADS * SEQ, 128, 0, stream>>>(scores, SEQ);

  // 12) o = P @ V (B is (K,N) slice of vv), written back into (B,L,H*dk); N=32 -> NT=2
  {
    GemmArgs g{};
    g.A = scores; g.lda = SEQ; g.sA1 = (long)N_HEADS * SEQ * SEQ; g.sA2 = (long)SEQ * SEQ;
    g.Bm = vv; g.ldb = D_MODEL; g.sB1 = (long)SEQ * D_MODEL; g.sB2 = DK;
    g.C = o; g.ldc = D_MODEL; g.sC1 = (long)SEQ * D_MODEL; g.sC2 = DK;
    g.M = SEQ; g.N = DK; g.K = SEQ; g.H = N_HEADS;
    g.alpha = 1.0f; g.bias = nullptr; g.resid = nullptr; g.ldr = D_MODEL;
    g.mask = nullptr; g.maskStride = 0;
    wmma_gemm_kernel<BL_NN, EP_NONE, 2><<<dim3(1, SEQ / 16, BATCH * N_HEADS), blk32, 0, stream>>>(g);
  }

  // 13) attention out proj + bias + residual h1 -> h2
  {
    GemmArgs g = mkargs(o, D_MODEL, wo_w, D_MODEL, h2, D_MODEL, TOKENS, D_MODEL, D_MODEL);
    g.bias = wo_b; g.resid = h1; g.ldr = D_MODEL;
    wmma_gemm_kernel<BL_NT, EP_BIAS_RESID, 4><<<dim3(D_MODEL / 64, TOKENS / 16, 1), blk32, 0, stream>>>(g);
  }

  // 14) norm3
  layernorm_kernel<<<TOKENS, D_MODEL, 0, stream>>>(h2, n3a, n3b, hn3);

  // 15) FFN up + relu
  {
    GemmArgs g = mkargs(hn3, D_MODEL, w1_w, D_MODEL, ff1, D_FF, TOKENS, D_FF, D_MODEL);
    g.bias = w1_b;
    wmma_gemm_kernel<BL_NT, EP_BIAS_RELU, 4><<<dim3(D_FF / 64, TOKENS / 16, 1), blk32, 0, stream>>>(g);
  }

  // 16) FFN down + bias + residual h2 -> out
  {
    GemmArgs g = mkargs(ff1, D_FF, w2_w, D_FF, out, D_MODEL, TOKENS, D_MODEL, D_FF);
    g.bias = w2_b; g.resid = h2; g.ldr = D_MODEL;
    wmma_gemm_kernel<BL_NT, EP_BIAS_RESID, 4><<<dim3(D_MODEL / 64, TOKENS / 16, 1), blk32, 0, stream>>>(g);
  }
}